// ICP_5196910428624
// MI455X (gfx1250) — compile-verified
//
#include <hip/hip_runtime.h>
#include <math.h>

typedef __attribute__((ext_vector_type(2))) float v2f;
typedef __attribute__((ext_vector_type(8))) float v8f;

#define NB 4
#define NPTS 4096
#define TOTPTS (NB * NPTS)       // 16384
#define NTILES (NPTS / 16)       // 256 tiles per batch
#define NPARTS (NB * NTILES)     // 1024 partial sums

// ws layout (in floats)
#define WS_ERR   0
#define WS_DONE  1
#define WS_APPLY 2
#define WS_RT    16              // 4 batches x 16 (R[9], t[3])
#define WS_COV   96              // 4 batches x 16 (S[9], sp[3], sq[3])
#define WS_PARTS 192             // 1024 NN partial sums
#define WS_TEMP  1280            // 4*4096*3 floats (current temppc)

// ---------------- init: temppc = newpc, clear state ----------------
__global__ __launch_bounds__(256) void icp_init(const float* __restrict__ newpc,
                                                float* __restrict__ ws) {
    int g = blockIdx.x * 256 + threadIdx.x;          // 16384 threads, 3 floats each
    float* temp = ws + WS_TEMP;
    temp[g]             = newpc[g];
    temp[g + TOTPTS]    = newpc[g + TOTPTS];
    temp[g + 2*TOTPTS]  = newpc[g + 2*TOTPTS];
    if (g == 0) { ws[WS_ERR] = 0.0f; ws[WS_DONE] = 0.0f; ws[WS_APPLY] = 0.0f; }
}

// ---------------- NN min-distance via WMMA f32 16x16x4 ----------------
// Grid: 128 blocks (4 batches x 32), 256 threads (8 waves). Wave w owns one
// 16-row origin tile; block stages the whole temp cloud of its batch in LDS.
__global__ __launch_bounds__(256) void icp_nn(const float* __restrict__ origin,
                                              float* __restrict__ ws) {
    __shared__ float4 sT[NPTS];                      // (x,y,z,|p|^2) : 64KB
    const int b   = blockIdx.x >> 5;
    const int tb  = blockIdx.x & 31;
    const int tid = threadIdx.x;

    const float* temp = ws + WS_TEMP + (size_t)b * NPTS * 3;
    for (int k = tid; k < NPTS; k += 256) {
        float x = temp[k*3+0], y = temp[k*3+1], z = temp[k*3+2];
        sT[k] = make_float4(x, y, z, x*x + y*y + z*z);
    }
    __syncthreads();

    const int wave = tid >> 5;
    const int lane = tid & 31;
    const int half = lane >> 4;                      // 0: K=0,1 / rows 0-7 ; 1: K=2,3 / rows 8-15
    const int l16  = lane & 15;
    const int tile = tb * 8 + wave;                  // origin tile within batch
    const int row  = tile * 16 + l16;

    const float* op = origin + ((size_t)b * NPTS + row) * 3;
    const float ox = op[0], oy = op[1], oz = op[2];
    const float no_own = ox*ox + oy*oy + oz*oz;

    // A fragment (16x4 f32): lane<16 -> (K0,K1)=(x,y); lane>=16 -> (K2,K3)=(z,0)
    v2f afrag;
    afrag[0] = half ? oz : ox;
    afrag[1] = half ? 0.0f : oy;

    v8f czero = {0.f,0.f,0.f,0.f,0.f,0.f,0.f,0.f};
    float minv[8];
    #pragma unroll
    for (int v = 0; v < 8; ++v) minv[v] = 3.0e38f;

    #pragma unroll 2
    for (int j = 0; j < NTILES; ++j) {
        float4 p = sT[j * 16 + l16];
        // B fragment (4x16 f32): same per-lane construction as A
        v2f bfrag;
        bfrag[0] = half ? p.z : p.x;
        bfrag[1] = half ? 0.0f : p.y;
        v8f acc = __builtin_amdgcn_wmma_f32_16x16x4_f32(
            false, afrag, false, bfrag, (short)0, czero, false, false);
        const float ntc = p.w;                       // |temp_col|^2 for this lane's column
        #pragma unroll
        for (int v = 0; v < 8; ++v) {
            float cand = __builtin_fmaf(-2.0f, acc[v], ntc);  // |t|^2 - 2*dot
            minv[v] = fminf(minv[v], cand);
        }
    }

    // add |origin_row|^2 (row of acc[v] = v + 8*half; its norm lives in lane v+24*half)
    #pragma unroll
    for (int v = 0; v < 8; ++v) {
        float no_v = __shfl(no_own, v + half * 24, 32);
        minv[v] += no_v;
    }
    // min across the 16 columns of each half
    #pragma unroll
    for (int v = 0; v < 8; ++v) {
        float m = minv[v];
        m = fminf(m, __shfl_xor(m, 1, 32));
        m = fminf(m, __shfl_xor(m, 2, 32));
        m = fminf(m, __shfl_xor(m, 4, 32));
        m = fminf(m, __shfl_xor(m, 8, 32));
        minv[v] = m;
    }
    float s = 0.0f;
    #pragma unroll
    for (int v = 0; v < 8; ++v) s += sqrtf(fmaxf(minv[v], 0.0f));
    s += __shfl_xor(s, 16, 32);                      // combine row groups 0-7 / 8-15
    if (lane == 0) ws[WS_PARTS + b * NTILES + tile] = s;
}

// ---------------- covariance: S = sum p q^T, sp = sum p, sq = sum q ----------------
__global__ __launch_bounds__(256) void icp_cov(const float* __restrict__ P,
                                               const float* __restrict__ Q,
                                               float* __restrict__ ws) {
    const int b = blockIdx.x, tid = threadIdx.x;
    const float* Pb = P + (size_t)b * NPTS * 3;
    const float* Qb = Q + (size_t)b * NPTS * 3;
    float a[15];
    #pragma unroll
    for (int v = 0; v < 15; ++v) a[v] = 0.0f;
    for (int n = tid; n < NPTS; n += 256) {
        float px = Pb[n*3], py = Pb[n*3+1], pz = Pb[n*3+2];
        float qx = Qb[n*3], qy = Qb[n*3+1], qz = Qb[n*3+2];
        a[0] += px*qx; a[1] += px*qy; a[2] += px*qz;
        a[3] += py*qx; a[4] += py*qy; a[5] += py*qz;
        a[6] += pz*qx; a[7] += pz*qy; a[8] += pz*qz;
        a[9] += px; a[10] += py; a[11] += pz;
        a[12] += qx; a[13] += qy; a[14] += qz;
    }
    __shared__ float red[256];
    for (int v = 0; v < 15; ++v) {
        red[tid] = a[v];
        __syncthreads();
        for (int off = 128; off > 0; off >>= 1) {
            if (tid < off) red[tid] += red[tid + off];
            __syncthreads();
        }
        if (tid == 0) ws[WS_COV + b * 16 + v] = red[0];
        __syncthreads();
    }
}

// ---------------- 3x3 Kabsch (SVD via cyclic Jacobi on H^T H) ----------------
__device__ inline float det3(const float M[3][3]) {
    return M[0][0]*(M[1][1]*M[2][2]-M[1][2]*M[2][1])
         - M[0][1]*(M[1][0]*M[2][2]-M[1][2]*M[2][0])
         + M[0][2]*(M[1][0]*M[2][1]-M[1][1]*M[2][0]);
}

__device__ void kabsch3(const float H[3][3], float R[3][3]) {
    float A[3][3];
    for (int i = 0; i < 3; ++i)
        for (int j = 0; j < 3; ++j)
            A[i][j] = H[0][i]*H[0][j] + H[1][i]*H[1][j] + H[2][i]*H[2][j];
    float V[3][3] = {{1,0,0},{0,1,0},{0,0,1}};
    for (int sweep = 0; sweep < 10; ++sweep) {
        for (int pq = 0; pq < 3; ++pq) {
            const int p = (pq == 2) ? 1 : 0;
            const int q = (pq == 0) ? 1 : 2;
            float apq = A[p][q];
            if (fabsf(apq) < 1e-30f) continue;
            float theta = (A[q][q] - A[p][p]) / (2.0f * apq);
            float t = copysignf(1.0f, theta) / (fabsf(theta) + sqrtf(theta*theta + 1.0f));
            float c = 1.0f / sqrtf(t*t + 1.0f);
            float s = t * c;
            for (int k = 0; k < 3; ++k) { float u=A[k][p], w=A[k][q]; A[k][p]=c*u-s*w; A[k][q]=s*u+c*w; }
            for (int k = 0; k < 3; ++k) { float u=A[p][k], w=A[q][k]; A[p][k]=c*u-s*w; A[q][k]=s*u+c*w; }
            for (int k = 0; k < 3; ++k) { float u=V[k][p], w=V[k][q]; V[k][p]=c*u-s*w; V[k][q]=s*u+c*w; }
        }
    }
    float lam[3] = {A[0][0], A[1][1], A[2][2]};
    for (int i = 0; i < 2; ++i)
        for (int j = i + 1; j < 3; ++j)
            if (lam[j] > lam[i]) {
                float tl = lam[i]; lam[i] = lam[j]; lam[j] = tl;
                for (int k = 0; k < 3; ++k) { float tv=V[k][i]; V[k][i]=V[k][j]; V[k][j]=tv; }
            }
    float U[3][3];
    for (int c = 0; c < 3; ++c) {
        float u0 = H[0][0]*V[0][c] + H[0][1]*V[1][c] + H[0][2]*V[2][c];
        float u1 = H[1][0]*V[0][c] + H[1][1]*V[1][c] + H[1][2]*V[2][c];
        float u2 = H[2][0]*V[0][c] + H[2][1]*V[1][c] + H[2][2]*V[2][c];
        float nrm = sqrtf(u0*u0 + u1*u1 + u2*u2);
        if (nrm > 1e-20f) { float inv = 1.0f/nrm; u0*=inv; u1*=inv; u2*=inv; }
        else if (c == 2) {
            u0 = U[1][0]*U[2][1] - U[2][0]*U[1][1];
            u1 = U[2][0]*U[0][1] - U[0][0]*U[2][1];
            u2 = U[0][0]*U[1][1] - U[1][0]*U[0][1];
        } else { u0 = (c == 0) ? 1.f : 0.f; u1 = (c == 1) ? 1.f : 0.f; u2 = 0.f; }
        U[0][c] = u0; U[1][c] = u1; U[2][c] = u2;
    }
    float sgn = (det3(V) * det3(U) < 0.0f) ? -1.0f : 1.0f;   // reflection fix on smallest sigma
    for (int i = 0; i < 3; ++i)
        for (int j = 0; j < 3; ++j)
            R[i][j] = V[i][0]*U[j][0] + V[i][1]*U[j][1] + sgn * V[i][2]*U[j][2];
}

__device__ inline void kabsch_from_cov(const float* cv, float R[3][3], float t[3]) {
    const float invN = 1.0f / (float)NPTS;
    float sp[3] = {cv[9], cv[10], cv[11]};
    float sq[3] = {cv[12], cv[13], cv[14]};
    float H[3][3];
    for (int i = 0; i < 3; ++i)
        for (int j = 0; j < 3; ++j)
            H[i][j] = cv[i*3+j] - sp[i]*sq[j]*invN;
    kabsch3(H, R);
    for (int i = 0; i < 3; ++i) {
        float c1_0 = sp[0]*invN, c1_1 = sp[1]*invN, c1_2 = sp[2]*invN;
        t[i] = sq[i]*invN - (R[i][0]*c1_0 + R[i][1]*c1_1 + R[i][2]*c1_2);
    }
}

// ---------------- per-iteration solve: err reduce + state + per-batch R,t ----------------
__global__ __launch_bounds__(256) void icp_solve(float* __restrict__ ws) {
    const int tid = threadIdx.x;
    __shared__ float red[256];
    float s = ws[WS_PARTS + tid] + ws[WS_PARTS + 256 + tid]
            + ws[WS_PARTS + 512 + tid] + ws[WS_PARTS + 768 + tid];
    red[tid] = s;
    __syncthreads();
    for (int off = 128; off > 0; off >>= 1) {
        if (tid < off) red[tid] += red[tid + off];
        __syncthreads();
    }
    if (tid == 0) {
        float errnew = red[0] * (1.0f / (float)TOTPTS);
        float err  = ws[WS_ERR];
        float done = ws[WS_DONE];
        float conv = (fabsf(err - errnew) < 1e-4f) ? 1.0f : 0.0f;
        ws[WS_APPLY] = (done != 0.0f) ? 0.0f : 1.0f;   // gated by OLD done (ref semantics)
        ws[WS_ERR]   = (done != 0.0f) ? err : errnew;
        ws[WS_DONE]  = fmaxf(done, conv);
    }
    if (tid < NB) {
        float R[3][3], t[3];
        kabsch_from_cov(ws + WS_COV + tid * 16, R, t);
        float* rt = ws + WS_RT + tid * 16;
        for (int i = 0; i < 3; ++i)
            for (int j = 0; j < 3; ++j) rt[i*3+j] = R[i][j];
        rt[9] = t[0]; rt[10] = t[1]; rt[11] = t[2];
    }
}

// ---------------- apply rigid transform (in place, gated) ----------------
__global__ __launch_bounds__(256) void icp_transform(float* __restrict__ ws) {
    if (ws[WS_APPLY] == 0.0f) return;
    const int g = blockIdx.x * 256 + threadIdx.x;     // 16384 points
    const int b = g >> 12;
    const float* rt = ws + WS_RT + b * 16;
    float* p = ws + WS_TEMP + (size_t)g * 3;
    float x = p[0], y = p[1], z = p[2];
    p[0] = rt[0]*x + rt[1]*y + rt[2]*z + rt[9];
    p[1] = rt[3]*x + rt[4]*y + rt[5]*z + rt[10];
    p[2] = rt[6]*x + rt[7]*y + rt[8]*z + rt[11];
}

// ---------------- final: [R | t] per batch to output ----------------
__global__ __launch_bounds__(32) void icp_final(const float* __restrict__ ws,
                                                float* __restrict__ out) {
    const int tid = threadIdx.x;
    if (tid >= NB) return;
    float R[3][3], t[3];
    kabsch_from_cov(ws + WS_COV + tid * 16, R, t);
    float* o = out + tid * 12;
    for (int r = 0; r < 3; ++r) {
        o[r*4+0] = R[r][0]; o[r*4+1] = R[r][1]; o[r*4+2] = R[r][2];
        o[r*4+3] = t[r];
    }
}

extern "C" void kernel_launch(void* const* d_in, const int* in_sizes, int n_in,
                              void* d_out, int out_size, void* d_ws, size_t ws_size,
                              hipStream_t stream) {
    const float* newpc    = (const float*)d_in[0];
    const float* originpc = (const float*)d_in[1];
    float* out = (float*)d_out;
    float* ws  = (float*)d_ws;
    (void)in_sizes; (void)n_in; (void)out_size; (void)ws_size;

    icp_init<<<TOTPTS/256, 256, 0, stream>>>(newpc, ws);

    for (int it = 0; it < 21; ++it) {                 // STEPLIM+1 scan steps
        icp_nn<<<NB * 32, 256, 0, stream>>>(originpc, ws);
        icp_cov<<<NB, 256, 0, stream>>>(ws + WS_TEMP, originpc, ws);  // P=temp, Q=origin
        icp_solve<<<1, 256, 0, stream>>>(ws);
        icp_transform<<<TOTPTS/256, 256, 0, stream>>>(ws);
    }
    // final transform: newpc -> converged cloud
    icp_cov<<<NB, 256, 0, stream>>>(newpc, ws + WS_TEMP, ws);         // P=newpc, Q=temp
    icp_final<<<1, 32, 0, stream>>>(ws, out);
}